// KmeansVectorQuantizer_55327768707656
// MI455X (gfx1250) — compile-verified
//
#include <hip/hip_runtime.h>
#include <hip/hip_bf16.h>

typedef __attribute__((ext_vector_type(2))) float v2f;
typedef __attribute__((ext_vector_type(8))) float v8f;

#define DIMC 512
#define GROUPS 2
#define VD 256           // per-group channel dim
#define NV 320           // codebook size
#define TLEN 2048
#define NB 16
#define BT (NB * TLEN)   // 32768 rows
#define MROWS 32         // rows per block (two 16-row WMMA M-subtiles)
#define APAD 258         // LDS row stride (even -> 8B aligned v2f, breaks 64-bank stride)

// workspace layout (float offsets)
#define WS_SUMS   0        // 32  (b+g)
#define WS_SUMSQ  32       // 32
#define WS_MU     64       // 32
#define WS_RSIG   96       // 32
#define WS_COUNTS 128      // 640 (g*320+v)
#define WS_LOSS   768      // 1
#define WS_ESQ    800      // 640 (g*320+v)
#define WS_WT     2048     // 2*128*256 float2 : wP[g][p][o] = {w[2p][o], w[2p+1][o]}
#define WS_EMBT   133120   // 2*128*320 float2 : eP[g][p][v] = {e[2p][v], e[2p+1][v]}

__device__ __forceinline__ v8f wmma_f32(v2f a, v2f b, v8f c) {
  // D = A(16x4) * B(4x16) + C(16x16), all f32.
  return __builtin_amdgcn_wmma_f32_16x16x4_f32(false, a, false, b, (short)0, c,
                                               false, false);
}

// Async LDS-DMA: one 8-byte element per lane, memory -> LDS, tracked by ASYNCcnt.
__device__ __forceinline__ void async_ld_b64(unsigned lds_byte_off,
                                             const float* gptr) {
  asm volatile("global_load_async_to_lds_b64 %0, %1, off"
               :: "v"(lds_byte_off), "v"(gptr)
               : "memory");
}
__device__ __forceinline__ void wait_async0() {
  asm volatile("s_wait_asynccnt 0" ::: "memory");
}

// Stage one 32x256 group-slice of x into LDS (row stride APAD) on the async path.
__device__ __forceinline__ void load_x_tile_async(const float* lds_base,
                                                  const float* __restrict__ x,
                                                  int row_base, int g, int tid) {
  const float* src = x + (size_t)row_base * DIMC + g * VD;
  const unsigned lds0 = (unsigned)(size_t)lds_base;   // low 32 bits = LDS offset
#pragma unroll
  for (int it = 0; it < 16; ++it) {
    int f2 = it * 256 + tid;     // 4096 float2 elements
    int m  = f2 >> 7;            // 0..31
    int c2 = (f2 & 127) * 2;     // even column
    async_ld_b64(lds0 + (unsigned)(m * APAD + c2) * 4u,
                 src + (size_t)m * DIMC + c2);
  }
  wait_async0();
}

// Per-wave conv GEMM: 32 rows x 32 out-cols (2x2 16x16 tiles); every B fragment
// feeds two WMMAs (two M-subtiles) -> 4 WMMA per (2 LDS + 2 VMEM) b64 loads.
__device__ __forceinline__ void conv_tile32(const float* lds, const float* __restrict__ wPg,
                                            int lane, int nb,
                                            v8f& a00, v8f& a01, v8f& a10, v8f& a11) {
  const int m  = lane & 15;
  const int kb = (lane >> 4) << 1;  // lanes 0-15: K+{0,1}; lanes 16-31: K+{2,3}
  const int n  = lane & 15;
  const int ph = lane >> 4;         // pair-index offset within K-quad
#pragma unroll 4
  for (int k0 = 0; k0 < VD; k0 += 4) {
    v2f aA = *(const v2f*)(lds + m * APAD + k0 + kb);
    v2f aB = *(const v2f*)(lds + (m + 16) * APAD + k0 + kb);
    const int pk = (k0 >> 1) + ph;  // K-pair index
    v2f b0 = *(const v2f*)(wPg + ((size_t)(pk * VD + nb + n) << 1));
    v2f b1 = *(const v2f*)(wPg + ((size_t)(pk * VD + nb + 16 + n) << 1));
    a00 = wmma_f32(aA, b0, a00);
    a01 = wmma_f32(aA, b1, a01);
    a10 = wmma_f32(aB, b0, a10);
    a11 = wmma_f32(aB, b1, a11);
  }
}

__global__ void k_zero(float* __restrict__ ws) {
  for (int i = threadIdx.x; i < 1440; i += 256) ws[i] = 0.f;
}

// wP[g][p][o] = { cw[g][o][2p], cw[g][o][2p+1] }
__global__ void k_prep_w(const float* __restrict__ cw, float* __restrict__ ws) {
  int i = blockIdx.x * 256 + threadIdx.x;     // 0..65535 (pairs)
  int g = i >> 15, rem = i & 32767;
  int p = rem >> 8, o = rem & 255;
  v2f v;
  v.x = cw[g * 65536 + o * 256 + 2 * p];
  v.y = cw[g * 65536 + o * 256 + 2 * p + 1];
  *(v2f*)(ws + WS_WT + g * 65536 + ((size_t)(p * 256 + o) << 1)) = v;
}

// eP[g][p][v] = { emb[v][g][2p], emb[v][g][2p+1] }
__global__ void k_prep_emb(const float* __restrict__ emb, float* __restrict__ ws) {
  int i = blockIdx.x * 256 + threadIdx.x;     // 0..81919 (pairs)
  if (i < GROUPS * 128 * NV) {
    int g = i / (128 * NV), rem = i % (128 * NV);
    int p = rem / NV, v = rem % NV;
    v2f e;
    e.x = emb[((size_t)v * GROUPS + g) * VD + 2 * p];
    e.y = emb[((size_t)v * GROUPS + g) * VD + 2 * p + 1];
    *(v2f*)(ws + WS_EMBT + g * (128 * NV * 2) + ((size_t)(p * NV + v) << 1)) = e;
  }
}

__global__ void k_prep_esq(const float* __restrict__ emb, float* __restrict__ ws) {
  int t = blockIdx.x * 256 + threadIdx.x;     // g*320+v
  if (t < GROUPS * NV) {
    int g = t / NV, v = t % NV;
    const float* e = emb + ((size_t)v * GROUPS + g) * VD;
    float s = 0.f;
    for (int d = 0; d < VD; ++d) s += e[d] * e[d];
    ws[WS_ESQ + t] = s;
  }
}

// Pass 1: conv GEMM, accumulate per-(b,g) sum / sumsq for GroupNorm.
__global__ void __launch_bounds__(256) k_conv_stats(const float* __restrict__ x,
                                                    float* __restrict__ ws) {
  __shared__ float lds[MROWS * APAD];
  __shared__ float red[2];
  const int tid = threadIdx.x, lane = tid & 31, wave = tid >> 5;
  const int g = blockIdx.y;
  const int row_base = blockIdx.x * MROWS;
  if (tid < 2) red[tid] = 0.f;
  load_x_tile_async(lds, x, row_base, g, tid);
  __syncthreads();

  const float* wPg = ws + WS_WT + (size_t)g * VD * VD;
  v8f a00 = {0.f,0.f,0.f,0.f,0.f,0.f,0.f,0.f};
  v8f a01 = a00, a10 = a00, a11 = a00;
  conv_tile32(lds, wPg, lane, wave * 32, a00, a01, a10, a11);

  float s = 0.f, sq = 0.f;
#pragma unroll
  for (int r = 0; r < 8; ++r) {
    float y0 = a00[r], y1 = a01[r], y2 = a10[r], y3 = a11[r];
    s += (y0 + y1) + (y2 + y3);
    sq += (y0 * y0 + y1 * y1) + (y2 * y2 + y3 * y3);
  }
  // explicit wave32 tree reduction, then one LDS atomic per wave
#pragma unroll
  for (int off = 16; off > 0; off >>= 1) {
    s  += __shfl_down(s, off, 32);
    sq += __shfl_down(sq, off, 32);
  }
  if (lane == 0) {
    atomicAdd(&red[0], s);
    atomicAdd(&red[1], sq);
  }
  __syncthreads();
  if (tid == 0) {
    int sidx = ((row_base * GROUPS) >> 12) + g;   // b+g  (b = row_base/2048)
    atomicAdd(&ws[WS_SUMS  + sidx], red[0]);
    atomicAdd(&ws[WS_SUMSQ + sidx], red[1]);
  }
}

__global__ void k_stats_final(float* __restrict__ ws) {
  int t = threadIdx.x;
  if (t < 32) {
    const float N = (float)(TLEN * VD);
    float mu  = ws[WS_SUMS + t] / N;
    float var = ws[WS_SUMSQ + t] / N - mu * mu;
    ws[WS_MU + t] = mu;
    ws[WS_RSIG + t] = rsqrtf(var + 1e-5f);
  }
}

// Pass 2: recompute conv, normalize -> ze in LDS, codebook GEMM + argmin,
// gather zq -> x_out, indices, counts, squared-error accumulation.
__global__ void __launch_bounds__(256) k_vq(const float* __restrict__ x,
                                            const float* __restrict__ gn_w,
                                            const float* __restrict__ gn_b,
                                            const float* __restrict__ emb,
                                            float* __restrict__ ws,
                                            float* __restrict__ out) {
  __shared__ float lds[MROWS * APAD];
  __shared__ unsigned long long best[MROWS];
  __shared__ int rowIdx[MROWS];
  __shared__ float lossRed;
  const int tid = threadIdx.x, lane = tid & 31, wave = tid >> 5;
  const int g = blockIdx.y;
  const int row_base = blockIdx.x * MROWS;
  const int sidx = ((row_base * GROUPS) >> 12) + g;

  if (tid < MROWS) best[tid] = ~0ull;
  if (tid == 0) lossRed = 0.f;
  load_x_tile_async(lds, x, row_base, g, tid);
  __syncthreads();

  const float* wPg = ws + WS_WT + (size_t)g * VD * VD;
  const int nb = wave * 32;
  const int m  = lane & 15;
  const int kb = (lane >> 4) << 1;
  const int n  = lane & 15;
  const int ph = lane >> 4;
  const int mrow = ph * 8;

  v8f a00 = {0.f,0.f,0.f,0.f,0.f,0.f,0.f,0.f};
  v8f a01 = a00, a10 = a00, a11 = a00;
  conv_tile32(lds, wPg, lane, nb, a00, a01, a10, a11);
  __syncthreads();   // all A-fragment reads of x tile done before overwrite

  // GroupNorm affine -> ze back into the same LDS tile
  const float mu = ws[WS_MU + sidx];
  const float rsig = ws[WS_RSIG + sidx];
  const int c0 = nb + n, c1 = nb + 16 + n;
  const float w0s = gn_w[g * VD + c0] * rsig, b0s = gn_b[g * VD + c0];
  const float w1s = gn_w[g * VD + c1] * rsig, b1s = gn_b[g * VD + c1];
#pragma unroll
  for (int r = 0; r < 8; ++r) {
    int mm = mrow + r;
    lds[mm * APAD + c0]        = (a00[r] - mu) * w0s + b0s;
    lds[mm * APAD + c1]        = (a01[r] - mu) * w1s + b1s;
    lds[(mm + 16) * APAD + c0] = (a10[r] - mu) * w0s + b0s;
    lds[(mm + 16) * APAD + c1] = (a11[r] - mu) * w1s + b1s;
  }
  __syncthreads();

  // dots = ze(32x256) x embT[g](256x320); argmin(||e||^2 - 2*dot) per row
  const float* ePg  = ws + WS_EMBT + (size_t)g * 128 * NV * 2;
  const float* esqg = ws + WS_ESQ + g * NV;
  for (int nt = wave; nt < NV / 16; nt += 8) {
    const int vb = nt * 16;
    v8f d0 = {0.f,0.f,0.f,0.f,0.f,0.f,0.f,0.f};
    v8f d1 = d0;
#pragma unroll 4
    for (int k0 = 0; k0 < VD; k0 += 4) {
      v2f aA = *(const v2f*)(lds + m * APAD + k0 + kb);
      v2f aB = *(const v2f*)(lds + (m + 16) * APAD + k0 + kb);
      const int pk = (k0 >> 1) + ph;
      v2f bf = *(const v2f*)(ePg + ((size_t)(pk * NV + vb + n) << 1));
      d0 = wmma_f32(aA, bf, d0);
      d1 = wmma_f32(aB, bf, d1);
    }
    const int v = vb + n;
    const float es = esqg[v];
#pragma unroll
    for (int r = 0; r < 8; ++r) {
      float s0 = es - 2.0f * d0[r];
      float s1 = es - 2.0f * d1[r];
      unsigned u0 = __float_as_uint(s0);
      unsigned u1 = __float_as_uint(s1);
      u0 = (u0 & 0x80000000u) ? ~u0 : (u0 | 0x80000000u);  // order-preserving map
      u1 = (u1 & 0x80000000u) ? ~u1 : (u1 | 0x80000000u);
      atomicMin(&best[mrow + r],      (((unsigned long long)u0) << 32) | (unsigned)v);
      atomicMin(&best[mrow + r + 16], (((unsigned long long)u1) << 32) | (unsigned)v);
    }
  }
  __syncthreads();

  if (tid < MROWS) {
    int iv = (int)(best[tid] & 0xFFFFFFFFu);
    rowIdx[tid] = iv;
    out[(size_t)BT * DIMC + (size_t)(row_base + tid) * GROUPS + g] = (float)iv;
    atomicAdd(&ws[WS_COUNTS + g * NV + iv], 1.0f);
  }
  __syncthreads();

  // x_out (forward value == zq) + squared-error accumulation
  const int mm2 = tid >> 3;            // 32 rows, 8 threads per row
  const int cs = (tid & 7) * 32;       // 32 cols per thread
  const int iv = rowIdx[mm2];
  const float* ev = emb + ((size_t)iv * GROUPS + g) * VD;
  float* orow = out + (size_t)(row_base + mm2) * DIMC + g * VD;
  float lsum = 0.f;
#pragma unroll
  for (int j = 0; j < 32; ++j) {
    int c = cs + j;
    float zqv = ev[c];
    float d = zqv - lds[mm2 * APAD + c];
    orow[c] = zqv;
    lsum += d * d;
  }
#pragma unroll
  for (int off = 16; off > 0; off >>= 1) lsum += __shfl_down(lsum, off, 32);
  if (lane == 0) atomicAdd(&lossRed, lsum);
  __syncthreads();
  if (tid == 0) atomicAdd(&ws[WS_LOSS], lossRed);
}

__global__ void k_final(const float* __restrict__ ws, float* __restrict__ out) {
  if (threadIdx.x == 0) {
    float ppl = 0.f;
    for (int g = 0; g < GROUPS; ++g) {
      float h = 0.f;
      for (int v = 0; v < NV; ++v) {
        float p = ws[WS_COUNTS + g * NV + v] * (1.0f / (float)BT);
        h += p * logf(p + 1e-7f);
      }
      ppl += expf(-h);
    }
    float lat = ws[WS_LOSS] / (float)((size_t)BT * DIMC);
    out[(size_t)BT * DIMC + (size_t)BT * GROUPS + 0] = ppl;
    out[(size_t)BT * DIMC + (size_t)BT * GROUPS + 1] = lat * (1.0f + 0.25f);
  }
}

extern "C" void kernel_launch(void* const* d_in, const int* in_sizes, int n_in,
                              void* d_out, int out_size, void* d_ws, size_t ws_size,
                              hipStream_t stream) {
  const float* x   = (const float*)d_in[0];
  const float* cw  = (const float*)d_in[1];
  const float* gnw = (const float*)d_in[2];
  const float* gnb = (const float*)d_in[3];
  const float* emb = (const float*)d_in[4];
  float* out = (float*)d_out;
  float* ws  = (float*)d_ws;

  k_zero<<<1, 256, 0, stream>>>(ws);
  k_prep_w<<<65536 / 256, 256, 0, stream>>>(cw, ws);
  k_prep_emb<<<(GROUPS * 128 * NV + 255) / 256, 256, 0, stream>>>(emb, ws);
  k_prep_esq<<<3, 256, 0, stream>>>(emb, ws);

  dim3 grid(BT / MROWS, GROUPS);
  k_conv_stats<<<grid, 256, 0, stream>>>(x, ws);
  k_stats_final<<<1, 32, 0, stream>>>(ws);
  k_vq<<<grid, 256, 0, stream>>>(x, gnw, gnb, emb, ws, out);
  k_final<<<1, 64, 0, stream>>>(ws, out);
}